// ROIAlign_74371653698123
// MI455X (gfx1250) — compile-verified
//
#include <hip/hip_runtime.h>
#include <hip/hip_bf16.h>
#include <stdint.h>

#define C_  256
#define H_  256
#define W_  256
#define PH_ 7
#define PW_ 7

typedef unsigned int u32x4 __attribute__((ext_vector_type(4)));
typedef int          i32x4 __attribute__((ext_vector_type(4)));
typedef int          i32x8 __attribute__((ext_vector_type(8)));

__device__ __forceinline__ int imin(int a, int b) { return a < b ? a : b; }
__device__ __forceinline__ int imax(int a, int b) { return a > b ? a : b; }

// Issue one TDM load of a 2(x) x 2(y) x 256(c) f32 tile from NCHW global into LDS.
// D# layout per CDNA5 ISA ch.8 (CDNA5 ISA).
__device__ __forceinline__ void tdm_load_tile(uint32_t lds_off, uint64_t gaddr) {
  u32x4 g0;
  g0.x = 1u;                                                   // count=1, user mode, no gather
  g0.y = lds_off;                                              // lds_addr (bytes)
  g0.z = (uint32_t)gaddr;                                      // global_addr[31:0]
  g0.w = (uint32_t)((gaddr >> 32) & 0x01FFFFFFull) | (2u << 30); // addr[56:32] | type=2
  i32x8 g1;
  g1[0] = (int)(2u << 16);                    // wg_mask=0, data_size=2 (4B), no flags
  g1[1] = (int)((uint32_t)W_ << 16);          // barrier_addr=0 | tensor_dim0.lo16 = W
  g1[2] = (int)((uint32_t)H_ << 16);          // tensor_dim0.hi16=0 | tensor_dim1.lo16 = H
  g1[3] = (int)(2u << 16);                    // tensor_dim1.hi16=0 | tile_dim0 = 2
  g1[4] = (int)(2u | ((uint32_t)C_ << 16));   // tile_dim1 = 2 | tile_dim2 = C
  g1[5] = (int)W_;                            // tensor_dim0_stride.lo32 = W (y step)
  g1[6] = 0;                                  // dim0_stride.hi16=0 | dim1_stride.lo16 (65536 -> 0)
  g1[7] = (int)((uint32_t)(H_ * W_) >> 16);   // dim1_stride[47:16] = 1 (c step = H*W)
  i32x4 g2; g2[0] = C_; g2[1] = 0; g2[2] = 0; g2[3] = 0;  // tensor_dim2 = C; no dim3
  i32x4 g3; g3[0] = 0;  g3[1] = 0; g3[2] = 0; g3[3] = 0;  // no dim4
#if defined(__clang_major__) && (__clang_major__ >= 23)
  i32x8 g4 = {0, 0, 0, 0, 0, 0, 0, 0};
  __builtin_amdgcn_tensor_load_to_lds(g0, g1, g2, g3, g4, 0);
#else
  __builtin_amdgcn_tensor_load_to_lds(g0, g1, g2, g3, 0);
#endif
}

struct PointParams {
  int   tx, ty;                 // clamped 2x2 tile origin (always in-bounds)
  float w00, w01, w10, w11;     // per-slot weights (masks folded in)
};

__device__ __forceinline__ PointParams point_params(float xmn, float ymn,
                                                    float xmx, float ymx,
                                                    int ph, int pw) {
  const float Wf = (float)W_, Hf = (float)H_;
  float xs = ((float)pw + 0.5f) * (2.0f / (float)PW_) - 1.0f;
  float ys = ((float)ph + 0.5f) * (2.0f / (float)PH_) - 1.0f;
  float gx = (xmx - xmn) * xs + (xmn * (2.0f / Wf) - 1.0f);
  float gy = (ymx - ymn) * ys + (ymn * (2.0f / Hf) - 1.0f);
  float ix = ((gx + 1.0f) * Wf - 1.0f) * 0.5f;
  float iy = ((gy + 1.0f) * Hf - 1.0f) * 0.5f;
  float x0f = floorf(ix), y0f = floorf(iy);
  float wx1 = ix - x0f, wx0 = 1.0f - wx1;
  float wy1 = iy - y0f, wy0 = 1.0f - wy1;
  int x0 = (int)x0f, y0 = (int)y0f;
  int x1 = x0 + 1,   y1 = y0 + 1;
  float mx0 = (x0 >= 0 && x0 < W_) ? 1.0f : 0.0f;
  float mx1 = (x1 >= 0 && x1 < W_) ? 1.0f : 0.0f;
  float my0 = (y0 >= 0 && y0 < H_) ? 1.0f : 0.0f;
  float my1 = (y1 >= 0 && y1 < H_) ? 1.0f : 0.0f;
  int tx = imin(imax(x0, 0), W_ - 2);
  int ty = imin(imax(y0, 0), H_ - 2);
  // clamped tap coords land inside the 2x2 tile; accumulate masked weights per slot
  int cx0 = imin(imax(x0, 0), W_ - 1) - tx;
  int cx1 = imin(imax(x1, 0), W_ - 1) - tx;
  int cy0 = imin(imax(y0, 0), H_ - 1) - ty;
  int cy1 = imin(imax(y1, 0), H_ - 1) - ty;
  float a0 = wx0 * mx0, a1 = wx1 * mx1;
  float b0 = wy0 * my0, b1 = wy1 * my1;
  float sx0 = (cx0 == 0 ? a0 : 0.0f) + (cx1 == 0 ? a1 : 0.0f);
  float sx1 = (cx0 == 1 ? a0 : 0.0f) + (cx1 == 1 ? a1 : 0.0f);
  float sy0 = (cy0 == 0 ? b0 : 0.0f) + (cy1 == 0 ? b1 : 0.0f);
  float sy1 = (cy0 == 1 ? b0 : 0.0f) + (cy1 == 1 ? b1 : 0.0f);
  PointParams r;
  r.tx = tx; r.ty = ty;
  r.w00 = sy0 * sx0; r.w01 = sy0 * sx1;
  r.w10 = sy1 * sx0; r.w11 = sy1 * sx1;
  return r;
}

__global__ __launch_bounds__(C_) void roi_align_tdm_kernel(
    const float* __restrict__ fmap, const float* __restrict__ rois,
    float* __restrict__ out) {
  // TDM writes each tile channel-major: offset = (c*4 + yy*2 + xx) floats
  __shared__ __align__(16) float tiles[PW_][4 * C_];   // 7 x 4KB = 28KB
  __shared__ __align__(16) float wts[PW_][4];          // per-pw slot weights

  const int n  = blockIdx.x / PH_;
  const int ph = blockIdx.x % PH_;

  if (threadIdx.x < 32u) {
    const float sc = 0.25f;
    const float xmn = rois[n * 5 + 0] * sc;
    const float ymn = rois[n * 5 + 1] * sc;
    const float xmx = rois[n * 5 + 2] * sc;
    const float ymx = rois[n * 5 + 3] * sc;
    const int lane = (int)threadIdx.x;

    // Lane l computes point params for pw = l (lanes 7..31 compute harmless junk).
    PointParams pp = point_params(xmn, ymn, xmx, ymx, ph, lane);
    if (lane < PW_) {
      float4 w4 = make_float4(pp.w00, pp.w01, pp.w10, pp.w11);
      *reinterpret_cast<float4*>(&wts[lane][0]) = w4;   // ds_store_b128
    }
    int off_lane = pp.ty * W_ + pp.tx;

    // generic LDS pointer low 32 bits == wave-relative LDS byte offset
    const uint32_t lds0 = (uint32_t)(uintptr_t)&tiles[0][0];
#pragma unroll
    for (int pw = 0; pw < PW_; ++pw) {
      int off = __builtin_amdgcn_readlane(off_lane, pw);   // lane pw's tile origin -> SGPR
      uint64_t ga = (uint64_t)(uintptr_t)fmap + (uint64_t)(uint32_t)off * 4ull;
      tdm_load_tile(lds0 + (uint32_t)pw * (4u * C_ * 4u), ga);
    }
    __builtin_amdgcn_s_wait_tensorcnt(0);
  }
  __syncthreads();
  asm volatile("" ::: "memory");   // LDS tiles were written by TDM, not visible to compiler

  const int c = threadIdx.x;
  float res[PW_];
#pragma unroll
  for (int pw = 0; pw < PW_; ++pw) {
    const float4 w = *reinterpret_cast<const float4*>(&wts[pw][0]);      // LDS broadcast
    const float4 v = *reinterpret_cast<const float4*>(&tiles[pw][c * 4]); // 4 taps, b128
    res[pw] = w.x * v.x + w.y * v.y + w.z * v.z + w.w * v.w;
  }
  float* o = out + (((size_t)n * C_ + c) * PH_ + ph) * PW_;
#pragma unroll
  for (int pw = 0; pw < PW_; ++pw) o[pw] = res[pw];
}

extern "C" void kernel_launch(void* const* d_in, const int* in_sizes, int n_in,
                              void* d_out, int out_size, void* d_ws, size_t ws_size,
                              hipStream_t stream) {
  const float* fmap = (const float*)d_in[0];
  const float* rois = (const float*)d_in[1];
  float* out = (float*)d_out;
  const int N = in_sizes[1] / 5;
  dim3 grid((unsigned)(N * PH_));
  roi_align_tdm_kernel<<<grid, C_, 0, stream>>>(fmap, rois, out);
}